// ShrinkTM_85564338471563
// MI455X (gfx1250) — compile-verified
//
#include <hip/hip_runtime.h>
#include <math.h>

#define NR 80      // n (rows per batch)
#define MM 30      // max_m
#define MMP 32     // padded K for WMMA (cols 30,31 zero)

typedef __attribute__((ext_vector_type(2))) float v2f;
typedef __attribute__((ext_vector_type(8))) float v8f;

__device__ __forceinline__ float matern15f(float d) {
    const float S3 = 1.7320508075688772f;
    return (1.0f + S3 * d) * expf(-S3 * d);
}

// ---------------------------------------------------------------------------
// Kernel 1: per-batch 30x30 Vecchia weights (Cholesky solve), accumulate
// factors^2 into global accumulator for the m_par truncation decision.
// One wave (32 threads) per batch.
// ---------------------------------------------------------------------------
__global__ void __launch_bounds__(32)
vecchia_factors_kernel(const float* __restrict__ locs,
                       const int*   __restrict__ nn,
                       const float* __restrict__ log_ls,
                       float*       __restrict__ factors,   // [N, 32]
                       float*       __restrict__ acc,       // [32], pre-zeroed
                       int N)
{
    __shared__ float S[MM][MM + 1];
    __shared__ float Ll[MM][2];
    __shared__ float msk[MM];
    __shared__ float bv[MM];

    const int i = blockIdx.x;
    const int t = threadIdx.x;
    const float ls = expf(log_ls[0]);
    const int nvalid = (i < MM) ? i : MM;

    if (t < MM) {
        int j = nn[i * MM + t];
        Ll[t][0] = locs[2 * j + 0];
        Ll[t][1] = locs[2 * j + 1];
        msk[t] = (t < nvalid) ? 1.0f : 0.0f;
    }
    __syncthreads();

    const float lx = locs[2 * i + 0];
    const float ly = locs[2 * i + 1];
    if (t < MM) {
        float mr = msk[t];
        for (int c = 0; c < MM; ++c) {
            float dx = Ll[t][0] - Ll[c][0];
            float dy = Ll[t][1] - Ll[c][1];
            float d  = sqrtf(fmaxf(dx * dx + dy * dy, 1e-30f)) / ls;
            float v  = matern15f(d) * mr * msk[c];
            if (c == t) v += mr * 1e-12f + (1.0f - mr);   // nugget / identity pad
            S[t][c] = v;
        }
        float dx = lx - Ll[t][0];
        float dy = ly - Ll[t][1];
        float d  = sqrtf(fmaxf(dx * dx + dy * dy, 1e-30f)) / ls;
        bv[t] = matern15f(d) * mr;                        // K12 * mask
    }
    __syncthreads();

    // in-place Cholesky (lower), column-parallel right-looking
    for (int k = 0; k < MM; ++k) {
        if (t == 0) S[k][k] = sqrtf(S[k][k]);
        __syncthreads();
        if (t > k && t < MM) S[t][k] /= S[k][k];
        __syncthreads();
        if (t > k && t < MM) {
            float lik = S[t][k];
            for (int j = k + 1; j <= t; ++j) S[t][j] -= lik * S[j][k];
        }
        __syncthreads();
    }
    // forward solve L y = b
    for (int k = 0; k < MM; ++k) {
        if (t == k) bv[k] /= S[k][k];
        __syncthreads();
        if (t > k && t < MM) bv[t] -= S[t][k] * bv[k];
        __syncthreads();
    }
    // backward solve L^T z = y
    for (int k = MM - 1; k >= 0; --k) {
        if (t == k) bv[k] /= S[k][k];
        __syncthreads();
        if (t < k) bv[t] -= S[k][t] * bv[k];
        __syncthreads();
    }

    if (t < MMP) {
        float f = (t < MM) ? bv[t] * msk[t] : 0.0f;
        factors[(size_t)i * MMP + t] = f;
        if (t < MM) atomicAdd(&acc[t], f * f);
    }
}

// ---------------------------------------------------------------------------
// Kernel 2: global truncation decision + scaling vector + likelihood constants.
// Single wave.
// ---------------------------------------------------------------------------
__global__ void __launch_bounds__(32)
vecchia_params_kernel(const float* __restrict__ acc,
                      const float* __restrict__ kernel_theta,
                      const float* __restrict__ nug_shrink,
                      float*       __restrict__ prm,
                      int N, int n)
{
    const int t = threadIdx.x;
    bool flag = false;
    if (t < MM) {
        float counts = (float)(N - 1 - t);               // #{i : i > t}
        float avg = acc[t] / counts;
        flag = (avg >= 1e-4f);
    }
    unsigned long long bal = __ballot(flag);
    int m_par = __popcll(bal);

    float th2 = kernel_theta[2], th3 = kernel_theta[3];
    float ls_k = expf(th2) * 1.7320508075688772f;        // sqrt(2*1.5)
    float se = 0.0f;
    if (t < MM) {
        float sc = powf((float)(t + 1), -expf(th3));
        if (sc < 0.01f) sc = 0.0f;                       // SCALE_THRESH
        se = sc / ls_k;
    }
    prm[t]      = se;                                    // scal_eff[0..31]
    prm[32 + t] = (t < m_par) ? 1.0f : 0.0f;             // colmask[0..31]
    if (t == 0) {
        float nm = 1.0f - 1.0f / (expf(nug_shrink[0]) + 1.0f); // nug_mult
        float alpha = 1.0f / (nm * nm) + 2.0f;
        float beta  = alpha - 1.0f;
        float alpha_post = alpha + 0.5f * (float)n;
        prm[64] = alpha_post;
        prm[65] = alpha * logf(beta) + lgammaf(alpha_post) - lgammaf(alpha);
        prm[66] = kernel_theta[0];
        prm[67] = kernel_theta[1];
        prm[68] = beta;
    }
}

// ---------------------------------------------------------------------------
// Kernel 3: per-batch 80x80 Gram via V_WMMA_F32_16X16X4_F32, Matern kernel,
// blocked 80x80 Cholesky (trailing SYRK update also via WMMA),
// triangular solve + loglik reduction.  One block (8 waves) per batch.
// ---------------------------------------------------------------------------
__global__ void __launch_bounds__(256)
vecchia_loglik_kernel(const float* __restrict__ response,  // [n, N]
                      const float* __restrict__ scales,    // [N]
                      const int*   __restrict__ nn,        // [N, MM]
                      const float* __restrict__ factors,   // [N, 32]
                      const float* __restrict__ prm,
                      float*       __restrict__ out,
                      int N)
{
    __shared__ float xs[NR][MMP];      // X (prev, then scaled)  80x32
    __shared__ float Gm[NR][NR + 1];   // Gram / kernel / Cholesky  80x81
    __shared__ float fac[MMP];
    __shared__ float sef[MMP];
    __shared__ float sqv[NR];
    __shared__ float rv[NR];

    const int i    = blockIdx.x;
    const int tid  = threadIdx.x;
    const int lane = tid & 31;
    const int wave = tid >> 5;
    const int nvalid = (i < MM) ? i : MM;

    if (tid < MMP) {
        sef[tid] = prm[tid];
        fac[tid] = factors[(size_t)i * MMP + tid] * prm[32 + tid];
    }
    // gather prev[s][m] = response[s, nn[i,m]] * mask
    for (int idx = tid; idx < NR * MMP; idx += 256) {
        int s = idx >> 5, m = idx & 31;
        float v = 0.0f;
        if (m < nvalid) v = response[(size_t)s * N + nn[i * MM + m]];
        xs[s][m] = v;
    }
    __syncthreads();

    // f_mean and residual
    if (tid < NR) {
        float fm = 0.0f;
        for (int m = 0; m < MM; ++m) fm += xs[tid][m] * fac[m];
        rv[tid] = response[(size_t)tid * N + i] - fm;
    }
    __syncthreads();

    // scale X in place: x = prev * scaling / ls_k  (cols 30,31 stay zero)
    for (int idx = tid; idx < NR * MMP; idx += 256) {
        int s = idx >> 5, m = idx & 31;
        xs[s][m] *= sef[m];
    }
    __syncthreads();
    if (tid < NR) {
        float ss = 0.0f;
        for (int m = 0; m < MM; ++m) { float v = xs[tid][m]; ss += v * v; }
        sqv[tid] = ss;
    }
    __syncthreads();

    const float th0 = prm[66], th1 = prm[67];
    const float sig2 = expf(2.0f * (th0 + th1 * logf(scales[i])));

    // ---- Gram G = X X^T via f32 WMMA, lower-triangular 16x16 tiles ----
    // A (16x4 f32): lane l<16 holds M=l, VGPR v holds K = v + 2*(lane>>4)
    // C (16x16 f32): VGPR r, lanes 0-15 -> M=r, lanes 16-31 -> M=r+8, N=lane&15
    const int h   = lane >> 4;
    const int l16 = lane & 15;
    for (int p = wave; p < 15; p += 8) {               // wave-uniform: EXEC all ones
        int tm = 0;
        while (((tm + 1) * (tm + 2)) / 2 <= p) ++tm;
        int tn = p - (tm * (tm + 1)) / 2;
        v8f c = {};
        for (int kk = 0; kk < MMP; kk += 4) {
            v2f a, b;
            a.x = xs[tm * 16 + l16][kk + 2 * h + 0];
            a.y = xs[tm * 16 + l16][kk + 2 * h + 1];
            b.x = xs[tn * 16 + l16][kk + 2 * h + 0];   // B = X^T tile (symmetric layout)
            b.y = xs[tn * 16 + l16][kk + 2 * h + 1];
            c = __builtin_amdgcn_wmma_f32_16x16x4_f32(
                    false, a, false, b, (short)0, c, false, false);
        }
#pragma unroll
        for (int r = 0; r < 8; ++r) {
            int row = tm * 16 + r + 8 * h;
            int col = tn * 16 + l16;
            float g  = c[r];
            float d2 = sqv[row] + sqv[col] - 2.0f * g;
            float val;
            if (d2 > 0.0f) val = matern15f(sqrtf(d2));
            else           val = 1.0f;                 // matern15(0)
            val = sig2 * val + ((row == col) ? 1.0f : 0.0f);
            if (i == 0) val = (row == col) ? 1.0f : 0.0f;  // batch-0 -> identity
            Gm[row][col] = val;
        }
    }
    __syncthreads();

    // ---- blocked 80x80 Cholesky (lower), 16-wide panels ----
    // trailing Schur update done as SYRK on the WMMA pipe.
    for (int kb = 0; kb < NR; kb += 16) {
        // 1) factor 16x16 diagonal block
        for (int k = kb; k < kb + 16; ++k) {
            if (tid == 0) Gm[k][k] = sqrtf(Gm[k][k]);
            __syncthreads();
            int r = k + 1 + tid;
            if (r < kb + 16) Gm[r][k] /= Gm[k][k];
            __syncthreads();
            if (r < kb + 16) {
                float lrk = Gm[r][k];
                for (int cc = k + 1; cc <= r; ++cc) Gm[r][cc] -= lrk * Gm[cc][k];
            }
            __syncthreads();
        }
        if (kb + 16 >= NR) break;

        // 2) panel triangular solve: L[r][kb:kb+16] = A[r][kb:kb+16] Ldd^-T
        {
            int r = kb + 16 + tid;
            if (r < NR) {
                for (int j = 0; j < 16; ++j) {
                    float s = Gm[r][kb + j];
                    for (int t2 = 0; t2 < j; ++t2)
                        s -= Gm[r][kb + t2] * Gm[kb + j][kb + t2];
                    Gm[r][kb + j] = s / Gm[kb + j][kb + j];
                }
            }
        }
        __syncthreads();

        // 3) trailing update A[trail] -= P P^T via WMMA (K=16),
        //    lower-triangular tiles only. Reads hit panel columns
        //    [kb, kb+16); writes hit columns >= kb+16 -> no overlap.
        int ntiles = (NR - kb - 16) >> 4;
        int npairs = (ntiles * (ntiles + 1)) / 2;
        for (int p = wave; p < npairs; p += 8) {       // wave-uniform
            int tm = 0;
            while (((tm + 1) * (tm + 2)) / 2 <= p) ++tm;
            int tn = p - (tm * (tm + 1)) / 2;
            int rb = kb + 16 + tm * 16;
            int cb = kb + 16 + tn * 16;
            v8f c = {};
            for (int kk = 0; kk < 16; kk += 4) {
                v2f a, b;
                a.x = Gm[rb + l16][kb + kk + 2 * h + 0];
                a.y = Gm[rb + l16][kb + kk + 2 * h + 1];
                b.x = Gm[cb + l16][kb + kk + 2 * h + 0];
                b.y = Gm[cb + l16][kb + kk + 2 * h + 1];
                c = __builtin_amdgcn_wmma_f32_16x16x4_f32(
                        false, a, false, b, (short)0, c, false, false);
            }
#pragma unroll
            for (int r = 0; r < 8; ++r) {
                int row = rb + r + 8 * h;
                int col = cb + l16;
                Gm[row][col] -= c[r];                  // f32 WMMA has no A/B neg
            }
        }
        __syncthreads();
    }

    // ---- forward solve L y = resid ----
    for (int k = 0; k < NR; ++k) {
        if (tid == 0) rv[k] /= Gm[k][k];
        __syncthreads();
        int r = k + 1 + tid;
        if (r < NR) rv[r] -= Gm[r][k] * rv[k];
        __syncthreads();
    }

    // ---- loglik contribution ----
    if (tid == 0) {
        float alpha_post = prm[64], cterm = prm[65], beta = prm[68];
        float logdet = 0.0f, ssq = 0.0f;
        for (int k = 0; k < NR; ++k) {
            logdet += logf(Gm[k][k]);
            float y = rv[k];
            ssq += y * y;
        }
        float beta_post = beta + 0.5f * ssq;
        float ll = -logdet - alpha_post * logf(beta_post) + cterm;
        atomicAdd(out, -ll);
    }
}

// ---------------------------------------------------------------------------
extern "C" void kernel_launch(void* const* d_in, const int* in_sizes, int n_in,
                              void* d_out, int out_size, void* d_ws, size_t ws_size,
                              hipStream_t stream)
{
    (void)n_in; (void)out_size; (void)ws_size;
    const float* response     = (const float*)d_in[0];
    const float* locs         = (const float*)d_in[1];
    const float* scales       = (const float*)d_in[2];
    const float* log_ls       = (const float*)d_in[3];
    const float* kernel_theta = (const float*)d_in[4];
    const float* nug_shrink   = (const float*)d_in[5];
    const int*   nn           = (const int*)d_in[6];

    const int N = in_sizes[2];            // 6144
    const int n = in_sizes[0] / N;        // 80

    float* factors = (float*)d_ws;                     // N*32 floats
    float* acc     = factors + (size_t)N * MMP;        // 32 floats
    float* prm     = acc + 32;                         // 96 floats

    hipMemsetAsync(d_out, 0, sizeof(float), stream);
    hipMemsetAsync(acc, 0, 32 * sizeof(float), stream);

    vecchia_factors_kernel<<<N, 32, 0, stream>>>(locs, nn, log_ls, factors, acc, N);
    vecchia_params_kernel<<<1, 32, 0, stream>>>(acc, kernel_theta, nug_shrink, prm, N, n);
    vecchia_loglik_kernel<<<N, 256, 0, stream>>>(response, scales, nn, factors, prm,
                                                 (float*)d_out, N);
}